// DIN_44976897523792
// MI455X (gfx1250) — compile-verified
//
#include <hip/hip_runtime.h>
#include <math.h>

// ---------------------------------------------------------------------------
// DIN attention, fused for MI455X (gfx1250, wave32).
//   h = u@(Wa+Wc) + i@(Wb-Wc) + (u*i)@Wd + b1   (W1 split into 4 ExH blocks)
//   DICE(h) -> w = .@W2 + b2 -> out[b,t,:] = sum_s w * u[b,s,:]
// Cross term (u*i)@Wd runs on V_WMMA_F32_16X16X4_F32 (exact fp32 matrix op).
// ---------------------------------------------------------------------------

typedef float v2f __attribute__((ext_vector_type(2)));
typedef float v8f __attribute__((ext_vector_type(8)));

#define B_   64
#define T_   200
#define TP_  50
#define E_   64
#define H_   36
#define NST  13          // ceil(T_/16)
#define LDP  68          // LDS pitch for 16x64 u tile (bank-conflict-free)

// workspace layout (float offsets)
#define WS_UA    0                          // B*T*H   = 460800
#define WS_IB    (WS_UA + B_*T_*H_)         // B*Tp*H  = 115200
#define WS_SUMS  (WS_IB + B_*TP_*H_)        // 72 (sum, sumsq)
#define WS_STATS (WS_SUMS + 80)             // 72 (mean, rstd)

__device__ inline v8f v8f_zero() {
  v8f v = {0.f,0.f,0.f,0.f,0.f,0.f,0.f,0.f};
  return v;
}

// B fragments for Wd = W1[3E : 4E, :]. B(k,n): lane = n + 16*(k/2), vgpr = k%2.
__device__ inline void din_load_bfrag(const float* __restrict__ W1, int lane,
                                      v2f bfrag[16][3]) {
  const int ncol = lane & 15;
  const int koff = (lane >> 4) * 2;
#pragma unroll
  for (int ks = 0; ks < 16; ++ks) {
    const int k0 = ks * 4 + koff;
#pragma unroll
    for (int nt = 0; nt < 3; ++nt) {
      const int n = nt * 16 + ncol;
      v2f bv = {0.f, 0.f};
      if (n < H_) {                                  // mask padded N columns
        bv.x = W1[(3 * E_ + k0)     * H_ + n];
        bv.y = W1[(3 * E_ + k0 + 1) * H_ + n];
      }
      bfrag[ks][nt] = bv;
    }
  }
}

// Stage one 16 x 64 tile of u[b, s0.. , :] into LDS (pitch LDP); pad rows -> 0.
__device__ inline void din_stage_u(const float* __restrict__ u, int b, int s0,
                                   int lane, float* lds_u) {
  const int row   = lane >> 1;
  const int cbase = (lane & 1) * 32;
  const int s     = s0 + row;
  const float* src = u + ((size_t)b * T_ + s) * E_;
#pragma unroll
  for (int j = 0; j < 8; ++j) {
    float4 v = make_float4(0.f, 0.f, 0.f, 0.f);
    if (s < T_) v = *(const float4*)(src + cbase + j * 4);
    *(float4*)(lds_u + row * LDP + cbase + j * 4) = v;
  }
}

// Cross-term tile: acc[nt] (16x16 f32) = (u_tile * it) @ Wd, 16 K-steps of
// V_WMMA_F32_16X16X4_F32 per N-tile. A(m,k): lane = m + 16*(k/2), vgpr = k%2.
__device__ inline void din_tile_mm(const float* __restrict__ lds_u,
                                   const float* __restrict__ lds_it,
                                   const v2f bfrag[16][3], int lane,
                                   v8f acc[3]) {
  acc[0] = v8f_zero(); acc[1] = v8f_zero(); acc[2] = v8f_zero();
  const int arow = lane & 15;
  const int koff = (lane >> 4) * 2;
#pragma unroll
  for (int ks = 0; ks < 16; ++ks) {
    const int k0 = ks * 4 + koff;
    const float2 uv = *(const float2*)(lds_u + arow * LDP + k0);
    const float2 iv = *(const float2*)(lds_it + k0);
    v2f a;
    a.x = uv.x * iv.x;
    a.y = uv.y * iv.y;
#pragma unroll
    for (int nt = 0; nt < 3; ++nt) {
      acc[nt] = __builtin_amdgcn_wmma_f32_16x16x4_f32(
          false, a, false, bfrag[ks][nt], (short)0, acc[nt], false, false);
    }
  }
}

// ---------------- pass 0: zero the stat accumulators -----------------------
__global__ void din_zero(float* __restrict__ sums) {
  const int i = threadIdx.x;
  if (i < 2 * H_) sums[i] = 0.f;
}

// ---------------- pass 1a: UA[b,s,:] = u @ (Wa + Wc) -----------------------
__global__ void din_ua(const float* __restrict__ u, const float* __restrict__ W1,
                       float* __restrict__ UA) {
  const int idx = blockIdx.x * blockDim.x + threadIdx.x;
  if (idx >= B_ * T_ * H_) return;
  const int row = idx / H_, ch = idx % H_;
  const float* ur = u + (size_t)row * E_;
  float s = 0.f;
#pragma unroll 4
  for (int e = 0; e < E_; ++e)
    s += ur[e] * (W1[e * H_ + ch] + W1[(2 * E_ + e) * H_ + ch]);
  UA[idx] = s;
}

// ---------------- pass 1b: IB[b,t,:] = i @ (Wb - Wc) + b1 ------------------
__global__ void din_ib(const float* __restrict__ it, const float* __restrict__ W1,
                       const float* __restrict__ b1, float* __restrict__ IB) {
  const int idx = blockIdx.x * blockDim.x + threadIdx.x;
  if (idx >= B_ * TP_ * H_) return;
  const int row = idx / H_, ch = idx % H_;
  const float* ir = it + (size_t)row * E_;
  float s = b1[ch];
#pragma unroll 4
  for (int e = 0; e < E_; ++e)
    s += ir[e] * (W1[(E_ + e) * H_ + ch] - W1[(2 * E_ + e) * H_ + ch]);
  IB[idx] = s;
}

// ---------------- pass 2: per-channel sum / sumsq of h ---------------------
__global__ __launch_bounds__(32)
void din_stats(const float* __restrict__ u, const float* __restrict__ it,
               const float* __restrict__ W1, const float* __restrict__ UA,
               const float* __restrict__ IB, float* __restrict__ sums) {
  __shared__ float lds_u[16 * LDP];
  __shared__ float lds_it[E_];
  __shared__ float lds_red[192];

  const int lane = threadIdx.x;
  const int bt = blockIdx.x;
  const int b = bt / TP_, t = bt % TP_;
  (void)t;

  if (lane < 16) {
    const float4 v = *(const float4*)(it + (size_t)bt * E_ + lane * 4);
    *(float4*)(lds_it + lane * 4) = v;
  }

  v2f bfrag[16][3];
  din_load_bfrag(W1, lane, bfrag);

  const int ncol = lane & 15;
  float ib[3];
#pragma unroll
  for (int nt = 0; nt < 3; ++nt) {
    const int n = nt * 16 + ncol;
    ib[nt] = (n < H_) ? IB[(size_t)bt * H_ + n] : 0.f;
  }

  float ssum[3] = {0.f, 0.f, 0.f}, ssq[3] = {0.f, 0.f, 0.f};
  const int mbase = (lane >> 4) * 8;

  for (int st = 0; st < NST; ++st) {
    __syncthreads();
    din_stage_u(u, b, st * 16, lane, lds_u);
    __syncthreads();

    v8f acc[3];
    din_tile_mm(lds_u, lds_it, bfrag, lane, acc);

    const int s0 = st * 16;
#pragma unroll
    for (int r = 0; r < 8; ++r) {
      const int s = s0 + mbase + r;
      if (s < T_) {                                  // padded rows excluded
#pragma unroll
        for (int nt = 0; nt < 3; ++nt) {
          const int n = nt * 16 + ncol;
          if (n < H_) {
            const float h = acc[nt][r] + UA[((size_t)b * T_ + s) * H_ + n] + ib[nt];
            ssum[nt] += h;
            ssq[nt]  += h * h;
          }
        }
      }
    }
  }

#pragma unroll
  for (int nt = 0; nt < 3; ++nt) {
    lds_red[nt * 32 + lane]      = ssum[nt];
    lds_red[96 + nt * 32 + lane] = ssq[nt];
  }
  __syncthreads();
  if (lane < 16) {
#pragma unroll
    for (int nt = 0; nt < 3; ++nt) {
      const int n = nt * 16 + lane;
      if (n < H_) {
        atomicAdd(&sums[n],       lds_red[nt * 32 + lane] + lds_red[nt * 32 + lane + 16]);
        atomicAdd(&sums[H_ + n],  lds_red[96 + nt * 32 + lane] + lds_red[96 + nt * 32 + lane + 16]);
      }
    }
  }
}

// ---------------- pass 3: mean / rstd --------------------------------------
__global__ void din_finalize_stats(const float* __restrict__ sums,
                                   float* __restrict__ stats) {
  const int n = threadIdx.x;
  if (n < H_) {
    const float invN = 1.0f / (float)(B_ * TP_ * T_);
    const float mean = sums[n] * invN;
    const float var  = sums[H_ + n] * invN - mean * mean;
    stats[n]       = mean;
    stats[H_ + n]  = rsqrtf(var + 0.001f);
  }
}

// ---------------- pass 4: recompute h, DICE, w, and out matvec -------------
__global__ __launch_bounds__(32)
void din_out(const float* __restrict__ u, const float* __restrict__ it,
             const float* __restrict__ W1, const float* __restrict__ UA,
             const float* __restrict__ IB, const float* __restrict__ stats,
             const float* __restrict__ alpha, const float* __restrict__ W2,
             const float* __restrict__ b2, float* __restrict__ out) {
  __shared__ float lds_u[16 * LDP];
  __shared__ float lds_it[E_];
  __shared__ float lds_w[16 * 17];
  __shared__ float lds_wrow[16];

  const int lane = threadIdx.x;
  const int bt = blockIdx.x;
  const int b = bt / TP_;

  if (lane < 16) {
    const float4 v = *(const float4*)(it + (size_t)bt * E_ + lane * 4);
    *(float4*)(lds_it + lane * 4) = v;
  }

  v2f bfrag[16][3];
  din_load_bfrag(W1, lane, bfrag);

  const int ncol = lane & 15;
  float ib[3], mn[3], rs[3], al[3], w2v[3];
#pragma unroll
  for (int nt = 0; nt < 3; ++nt) {
    const int n = nt * 16 + ncol;
    if (n < H_) {
      ib[nt]  = IB[(size_t)bt * H_ + n];
      mn[nt]  = stats[n];
      rs[nt]  = stats[H_ + n];
      al[nt]  = alpha[n];
      w2v[nt] = W2[n];
    } else {
      ib[nt] = 0.f; mn[nt] = 0.f; rs[nt] = 0.f; al[nt] = 0.f; w2v[nt] = 0.f;
    }
  }
  const float bias2 = b2[0];

  const int mbase = (lane >> 4) * 8;
  const int e0 = lane * 2;
  float2 oacc = make_float2(0.f, 0.f);

  for (int st = 0; st < NST; ++st) {
    __syncthreads();
    din_stage_u(u, b, st * 16, lane, lds_u);
    __syncthreads();

    v8f acc[3];
    din_tile_mm(lds_u, lds_it, bfrag, lane, acc);

    const int s0 = st * 16;
#pragma unroll
    for (int r = 0; r < 8; ++r) {
      const int m = mbase + r;
      const int s = s0 + m;
      float part = 0.f;
#pragma unroll
      for (int nt = 0; nt < 3; ++nt) {
        const float ua = ((nt * 16 + ncol) < H_ && s < T_)
                             ? UA[((size_t)b * T_ + s) * H_ + (nt * 16 + ncol)]
                             : 0.f;
        const float h = acc[nt][r] + ua + ib[nt];
        const float z = (h - mn[nt]) * rs[nt];
        const float p = 1.0f / (1.0f + __expf(-z));
        const float g = h * (al[nt] + p * (1.0f - al[nt]));    // DICE
        part += g * w2v[nt];                                   // h @ W2
      }
      lds_w[m * 17 + ncol] = part;
    }
    __syncthreads();
    if (lane < 16) {
      float wv = bias2;
#pragma unroll
      for (int c = 0; c < 16; ++c) wv += lds_w[lane * 17 + c];
      lds_wrow[lane] = wv;                                     // weight[b,t,s]
    }
    __syncthreads();
    // out[b,t,e] += sum_m w[s0+m] * u[b,s0+m,e]  (padded rows are zeros)
#pragma unroll
    for (int m = 0; m < 16; ++m) {
      const float wv = lds_wrow[m];
      oacc.x += wv * lds_u[m * LDP + e0];
      oacc.y += wv * lds_u[m * LDP + e0 + 1];
    }
  }

  float* o = out + (size_t)bt * E_;
  o[e0]     = oacc.x;
  o[e0 + 1] = oacc.y;
}

// ---------------------------------------------------------------------------
extern "C" void kernel_launch(void* const* d_in, const int* in_sizes, int n_in,
                              void* d_out, int out_size, void* d_ws,
                              size_t ws_size, hipStream_t stream) {
  const float* u     = (const float*)d_in[0];
  const float* items = (const float*)d_in[1];
  const float* W1    = (const float*)d_in[2];
  const float* b1    = (const float*)d_in[3];
  const float* alpha = (const float*)d_in[4];
  const float* W2    = (const float*)d_in[5];
  const float* b2    = (const float*)d_in[6];
  float* out = (float*)d_out;
  float* ws  = (float*)d_ws;

  float* UA    = ws + WS_UA;
  float* IB    = ws + WS_IB;
  float* sums  = ws + WS_SUMS;
  float* stats = ws + WS_STATS;

  din_zero<<<1, 128, 0, stream>>>(sums);
  din_ua<<<(B_ * T_ * H_ + 255) / 256, 256, 0, stream>>>(u, W1, UA);
  din_ib<<<(B_ * TP_ * H_ + 255) / 256, 256, 0, stream>>>(items, W1, b1, IB);
  din_stats<<<B_ * TP_, 32, 0, stream>>>(u, items, W1, UA, IB, sums);
  din_finalize_stats<<<1, 64, 0, stream>>>(sums, stats);
  din_out<<<B_ * TP_, 32, 0, stream>>>(u, items, W1, UA, IB, stats, alpha, W2,
                                       b2, out);
}